// AttEncoder_12850542150202
// MI455X (gfx1250) — compile-verified
//
#include <hip/hip_runtime.h>
#include <hip/hip_bf16.h>

typedef __attribute__((ext_vector_type(2))) float v2f;
typedef __attribute__((ext_vector_type(8))) float v8f;
typedef __attribute__((ext_vector_type(4))) int v4i;

#define KEY_DIM 128
#define VAL_DIM 64
#define CAT_DIM 192  // KEY_DIM + VAL_DIM

// ---------------- K0: zero row_sum + S accumulators ----------------
__global__ void zero_ws_kernel(float* __restrict__ p, long long n) {
    long long i = (long long)blockIdx.x * blockDim.x + threadIdx.x;
    long long stride = (long long)gridDim.x * blockDim.x;
    for (; i < n; i += stride) p[i] = 0.0f;
}

// ---------------- K1: per-edge attention score + row_sum -----------
// One wave (32 lanes) per edge. 256-dim dot: each lane does one float4
// of ent_feats[h]·a_w[0:128] and one float4 of att_feats[att]·a_w[128:256].
__global__ __launch_bounds__(256) void score_kernel(
    const int* __restrict__ triples, const float* __restrict__ ent_feats,
    const float* __restrict__ att_feats, const float* __restrict__ a_w,
    const float* __restrict__ a_b, float* __restrict__ scores,
    float* __restrict__ row_sum, int n_edges) {
  int lane = threadIdx.x & 31;
  int e = blockIdx.x * (blockDim.x >> 5) + (threadIdx.x >> 5);
  if (e >= n_edges) return;
  int h = triples[3 * e + 0];
  int att = triples[3 * e + 1];
  const float4* eh = (const float4*)(ent_feats + (long long)h * KEY_DIM);
  const float4* af = (const float4*)(att_feats + (long long)att * KEY_DIM);
  const float4* wlo = (const float4*)a_w;              // 32 x float4
  const float4* whi = (const float4*)(a_w + KEY_DIM);  // 32 x float4
  float4 x0 = eh[lane], w0 = wlo[lane];
  float4 x1 = af[lane], w1 = whi[lane];
  float d = x0.x * w0.x + x0.y * w0.y + x0.z * w0.z + x0.w * w0.w +
            x1.x * w1.x + x1.y * w1.y + x1.z * w1.z + x1.w * w1.w;
#pragma unroll
  for (int off = 16; off > 0; off >>= 1) d += __shfl_xor(d, off, 32);
  if (lane == 0) {
    float s = d + a_b[0];
    s = (s >= 0.0f) ? s : 0.2f * s;  // leaky_relu(0.2)
    s = expf(s);
    scores[e] = s;
    atomicAdd(&row_sum[h], s);
  }
}

// ---------------- K2: weighted concat-feature aggregation ----------
// One wave per edge; 192 components, 6 per lane (stride 32); atomic
// f32 adds land in L2 (S accumulator is L2-resident, 77 MB << 192 MB).
__global__ __launch_bounds__(256) void aggregate_kernel(
    const int* __restrict__ triples, const float* __restrict__ att_feats,
    const float* __restrict__ val_feats, const float* __restrict__ scores,
    const float* __restrict__ row_sum, float* __restrict__ S, int n_edges) {
  int lane = threadIdx.x & 31;
  int e = blockIdx.x * (blockDim.x >> 5) + (threadIdx.x >> 5);
  if (e >= n_edges) return;
  int h = triples[3 * e + 0];
  int att = triples[3 * e + 1];
  int val = triples[3 * e + 2];
  float p = scores[e] / row_sum[h];
  const float* af = att_feats + (long long)att * KEY_DIM;
  const float* vf = val_feats + (long long)val * VAL_DIM;
  float* dstr = S + (long long)h * CAT_DIM;
#pragma unroll
  for (int i = 0; i < 6; ++i) {
    int c = lane + 32 * i;
    float v = (c < KEY_DIM) ? af[c] : vf[c - KEY_DIM];
    atomicAdd(&dstr[c], p * v);
  }
}

// ---------------- K3: node GEMM (S @ W) via f32 WMMA + ELU ---------
// Block = 256 threads = 8 waves. Block owns a 16-node M-tile, staged
// into 12 KB LDS with gfx1250 async global->LDS b128 copies (ASYNCcnt),
// reused by all 8 waves; wave w owns output cols [16w,16w+16).
// K = 192 swept in 48 steps of V_WMMA_F32_16X16X4_F32 (full f32 precision).
__global__ __launch_bounds__(256) void gemm_elu_kernel(
    const float* __restrict__ S, const float* __restrict__ W,
    const float* __restrict__ ent_feats, float* __restrict__ out,
    int n_nodes) {
  __shared__ float ldsS[16 * CAT_DIM];  // 12 KB
  int m0 = blockIdx.x * 16;
  int nrows = n_nodes - m0;  // >= 16 except possibly last tile
  // Stage S tile: 16 rows x 192 f32 = 768 x b128, 3 per thread.
  const float4* src = (const float4*)(S + (long long)m0 * CAT_DIM);
  float4* dst = (float4*)ldsS;
  const float4 z4 = {0.0f, 0.0f, 0.0f, 0.0f};
  for (int i = threadIdx.x; i < (16 * CAT_DIM) / 4; i += blockDim.x) {
    int r = (i * 4) / CAT_DIM;
    if (r < nrows) {
#if defined(__gfx1250__) && \
    __has_builtin(__builtin_amdgcn_global_load_async_to_lds_b128)
      // Direct global->LDS async copy (no VGPR round-trip), ASYNCcnt-tracked.
      // Param 0: v4i in addrspace(1) (global src); param 1: v4i in
      // addrspace(3) (LDS dst); then imm offset, imm cpol.
      __builtin_amdgcn_global_load_async_to_lds_b128(
          (__attribute__((address_space(1))) v4i*)(src + i),
          (__attribute__((address_space(3))) v4i*)(dst + i),
          /*offset=*/0, /*cpol=*/0);
#else
      dst[i] = src[i];
#endif
    } else {
      dst[i] = z4;  // zero-fill tail rows (never taken: 100000 % 16 == 0)
    }
  }
#if defined(__gfx1250__) && \
    __has_builtin(__builtin_amdgcn_global_load_async_to_lds_b128)
#if __has_builtin(__builtin_amdgcn_s_wait_asynccnt)
  __builtin_amdgcn_s_wait_asynccnt(0);
#else
  asm volatile("s_wait_asynccnt 0x0" ::: "memory");
#endif
#endif
  __syncthreads();

  int wave = threadIdx.x >> 5;
  int lane = threadIdx.x & 31;
  int n0 = wave * 16;
  int row = lane & 15;         // A: M-row | B/C/D: N-col within tile
  int koff = (lane >> 4) * 2;  // K sub-offset: lanes 0-15 -> K+{0,1}, 16-31 -> K+{2,3}

  v8f c = {};
  const float* arow = ldsS + row * CAT_DIM;
#pragma unroll 4
  for (int k0 = 0; k0 < CAT_DIM; k0 += 4) {
    v2f a, b;
    a.x = arow[k0 + koff];
    a.y = arow[k0 + koff + 1];
    b.x = W[(k0 + koff) * KEY_DIM + n0 + row];
    b.y = W[(k0 + koff + 1) * KEY_DIM + n0 + row];
    // 8 args: (neg_a, A, neg_b, B, c_mod, C, reuse_a, reuse_b)
    c = __builtin_amdgcn_wmma_f32_16x16x4_f32(false, a, false, b, (short)0, c,
                                              false, false);
  }

  // C/D layout: VGPR i holds M = i + 8*(lane>=16), N = n0 + (lane&15)
  int col = n0 + row;
  int mhalf = (lane >> 4) * 8;
#pragma unroll
  for (int i = 0; i < 8; ++i) {
    int m = m0 + mhalf + i;
    if (m < n_nodes) {
      float x = c[i] + ent_feats[(long long)m * KEY_DIM + col];
      out[(long long)m * KEY_DIM + col] = (x > 0.0f) ? x : (expf(x) - 1.0f);
    }
  }
}

extern "C" void kernel_launch(void* const* d_in, const int* in_sizes, int n_in,
                              void* d_out, int out_size, void* d_ws,
                              size_t ws_size, hipStream_t stream) {
  const int* triples = (const int*)d_in[0];
  const float* ent_feats = (const float*)d_in[1];
  const float* att_feats = (const float*)d_in[2];
  const float* val_feats = (const float*)d_in[3];
  const float* a_w = (const float*)d_in[4];
  const float* a_b = (const float*)d_in[5];
  const float* W = (const float*)d_in[6];
  float* out = (float*)d_out;

  int n_edges = in_sizes[0] / 3;
  int n_nodes = in_sizes[1] / KEY_DIM;

  // Workspace layout: [scores: n_edges][row_sum: n_nodes][S: n_nodes*192]
  float* scores = (float*)d_ws;
  float* row_sum = scores + n_edges;
  float* S = row_sum + n_nodes;

  // K0: zero row_sum + S (contiguous) every call
  long long nzero = (long long)n_nodes * (CAT_DIM + 1);
  zero_ws_kernel<<<2048, 256, 0, stream>>>(row_sum, nzero);

  // K1 + K2: 8 waves (edges) per 256-thread block
  int edge_blocks = (n_edges + 7) / 8;
  score_kernel<<<edge_blocks, 256, 0, stream>>>(
      triples, ent_feats, att_feats, a_w, a_b, scores, row_sum, n_edges);
  aggregate_kernel<<<edge_blocks, 256, 0, stream>>>(
      triples, att_feats, val_feats, scores, row_sum, S, n_edges);

  // K3: one block per 16-node tile (100000 / 16 = 6250 exact)
  int gemm_blocks = (n_nodes + 15) / 16;
  gemm_elu_kernel<<<gemm_blocks, 256, 0, stream>>>(S, W, ent_feats, out,
                                                   n_nodes);
}